// GCNBlock_SEGPR_79396765434249
// MI455X (gfx1250) — compile-verified
//
#include <hip/hip_runtime.h>
#include <hip/hip_bf16.h>

typedef __attribute__((ext_vector_type(2))) float v2f;
typedef __attribute__((ext_vector_type(8))) float v8f;

#define CIN 256
#define COUT 256
#define CHALF 128
#define BN_EPS 1e-5f

// ---------------- init: deg = 1 (self loop), stats = 0 ----------------
__global__ void k_init(float* __restrict__ deg, float* __restrict__ stats, int N) {
    int i = blockIdx.x * blockDim.x + threadIdx.x;
    if (i < N)   deg[i] = 1.0f;
    if (i < 2 * COUT) stats[i] = 0.0f;
}

// ---------------- degree: deg[col[e]] += 1 ----------------
__global__ void k_degree(const int* __restrict__ col, float* __restrict__ deg, int E) {
    int e = blockIdx.x * blockDim.x + threadIdx.x;
    if (e < E) atomicAdd(&deg[col[e]], 1.0f);
}

// ---------------- dinv: deg -> rsqrt(deg), in place (deg >= 1 always) ----------------
__global__ void k_dinv(float* __restrict__ deg, int N) {
    int i = blockIdx.x * blockDim.x + threadIdx.x;
    if (i < N) deg[i] = rsqrtf(deg[i]);
}

// ---------------- WMMA GEMM: y[m][n] = dinv[m] * sum_k x[m][k] * W[n][k] ----------------
// One block = 16 waves; block handles a 16-row M tile; wave w handles N tile w (16 cols).
// Uses V_WMMA_F32_16X16X4_F32 (fp32 in, fp32 accum) -- exact arithmetic, no downcast.
__global__ __launch_bounds__(512)
void k_gemm_wmma(const float* __restrict__ x, const float* __restrict__ W,
                 const float* __restrict__ dinv, float* __restrict__ y, int N) {
    __shared__ float xs[16][CIN];   // 16 KB x-tile
    __shared__ float sdv[16];
    const int m0  = blockIdx.x * 16;
    const int tid = threadIdx.x;

    // stage x tile (coalesced: 512 threads x 8 floats)
    #pragma unroll
    for (int i = 0; i < 8; ++i) {
        int flat = i * 512 + tid;
        int r = flat >> 8, c = flat & 255;
        int m = m0 + r; if (m >= N) m = N - 1;
        xs[r][c] = x[(long)m * CIN + c];
    }
    if (tid < 16) {
        int m = m0 + tid;
        sdv[tid] = (m < N) ? dinv[m] : 0.0f;
    }
    __syncthreads();

    const int wave = tid >> 5;     // 0..15 -> N tile
    const int lane = tid & 31;
    const int l16  = lane & 15;
    const int hi   = lane >> 4;    // 0: K=0..1 / rows 0..7 ; 1: K=2..3 / rows 8..15
    const int n0   = wave * 16;
    const float* __restrict__ wrow = W + (long)(n0 + l16) * CIN;   // B[k][n] = W[n][k]

    v8f acc = {};
    for (int k = 0; k < CIN; k += 4) {
        v2f a, b;
        // A 16x4: lanes 0-15 VGPR0=K0, VGPR1=K1; lanes 16-31 VGPR0=K2, VGPR1=K3
        a.x = xs[l16][k + 2 * hi];
        a.y = xs[l16][k + 2 * hi + 1];
        // B 4x16: VGPR0 = row K(2*hi), VGPR1 = row K(2*hi+1), N striped over lanes 0-15
        float2 wv = *(const float2*)(wrow + k + 2 * hi);
        b.x = wv.x; b.y = wv.y;
        acc = __builtin_amdgcn_wmma_f32_16x16x4_f32(
            /*neg_a=*/false, a, /*neg_b=*/false, b,
            /*c_mod=*/(short)0, acc, /*reuse_a=*/false, /*reuse_b=*/false);
    }

    // C/D 16x16 f32: VGPR r -> row (r + 8*hi), col = lane&15 within N tile
    #pragma unroll
    for (int r = 0; r < 8; ++r) {
        int m = m0 + r + 8 * hi;
        if (m < N) y[(long)m * COUT + n0 + l16] = acc[r] * sdv[r + 8 * hi];
    }
}

// ---------------- copy: agg = y (self-loop contribution) ----------------
__global__ void k_copy4(const float4* __restrict__ src, float4* __restrict__ dst, long n4) {
    long i = (long)blockIdx.x * blockDim.x + threadIdx.x;
    if (i < n4) dst[i] = src[i];
}

// ---------------- scatter: agg[col[e]] += y[row[e]]  (64 threads/edge, float4 each) ----
__global__ void k_scatter(const int* __restrict__ src, const int* __restrict__ dst,
                          const float* __restrict__ y, float* __restrict__ agg, int E) {
    long t = (long)blockIdx.x * blockDim.x + threadIdx.x;
    long total = (long)E * 64;
    if (t >= total) return;
    int e  = (int)(t >> 6);
    int c4 = (int)(t & 63) * 4;
    int r = src[e], c = dst[e];
    float4 v = *(const float4*)(y + (long)r * COUT + c4);   // y resident in 192MB L2
    float* p = agg + (long)c * COUT + c4;                   // atomics resolve in L2
    atomicAdd(p + 0, v.x); atomicAdd(p + 1, v.y);
    atomicAdd(p + 2, v.z); atomicAdd(p + 3, v.w);
}

// ---------------- finalize: z = relu(dinv[n]*agg + b), in place; accumulate BN stats ----
#define ROWS_PER_BLK 40
__global__ __launch_bounds__(256)
void k_finalize(float* __restrict__ agg, const float* __restrict__ dinv,
                const float* __restrict__ bias, float* __restrict__ stats, int N) {
    int c = threadIdx.x;           // channel
    int base = blockIdx.x * ROWS_PER_BLK;
    float bc = bias[c];
    float s = 0.0f, ss = 0.0f;
    for (int r = 0; r < ROWS_PER_BLK; ++r) {
        int n = base + r;
        if (n >= N) break;
        float dv = dinv[n];
        long idx = (long)n * COUT + c;
        float z = fmaxf(agg[idx] * dv + bc, 0.0f);
        agg[idx] = z;
        s += z; ss += z * z;
    }
    atomicAdd(&stats[c], s);
    atomicAdd(&stats[COUT + c], ss);
}

// ---------------- BN affine params: scale = gamma*rsqrt(var+eps), shift = beta - mean*scale
__global__ void k_bnparam(const float* __restrict__ stats, const float* __restrict__ gamma,
                          const float* __restrict__ beta, float* __restrict__ bnp, int N) {
    int c = threadIdx.x;
    if (c >= COUT) return;
    float invN = 1.0f / (float)N;
    float mean = stats[c] * invN;
    float var  = stats[COUT + c] * invN - mean * mean;   // biased, as jnp.var
    float sc   = gamma[c] * rsqrtf(var + BN_EPS);
    bnp[c]        = sc;
    bnp[COUT + c] = beta[c] - mean * sc;
}

// ---------------- BN + MaxPool1d(2): out[n][k] = max(bn(z[n][2k]), bn(z[n][2k+1])) ------
__global__ void k_out(const float* __restrict__ z, const float* __restrict__ bnp,
                      float* __restrict__ out, int N) {
    long i = (long)blockIdx.x * blockDim.x + threadIdx.x;
    long total = (long)N * CHALF;
    if (i >= total) return;
    int n = (int)(i >> 7);
    int k = (int)(i & 127);
    float2 v = *(const float2*)(z + (long)n * COUT + 2 * k);
    float a = v.x * bnp[2 * k]     + bnp[COUT + 2 * k];
    float b = v.y * bnp[2 * k + 1] + bnp[COUT + 2 * k + 1];
    out[i] = fmaxf(a, b);
}

extern "C" void kernel_launch(void* const* d_in, const int* in_sizes, int n_in,
                              void* d_out, int out_size, void* d_ws, size_t ws_size,
                              hipStream_t stream) {
    const float* x      = (const float*)d_in[0];
    const int*   ei     = (const int*)d_in[1];     // [2][E]: row 0 = src, row 1 = dst
    const float* W      = (const float*)d_in[2];
    const float* bias   = (const float*)d_in[3];
    const float* gamma  = (const float*)d_in[4];
    const float* beta   = (const float*)d_in[5];
    float*       out    = (float*)d_out;

    const int N = in_sizes[0] / CIN;
    const int E = in_sizes[1] / 2;
    const int* esrc = ei;
    const int* edst = ei + E;

    // workspace layout
    char* ws = (char*)d_ws;
    size_t featBytes = (size_t)N * COUT * sizeof(float);
    float* y     = (float*)(ws);                              // [N,256]
    float* agg   = (float*)(ws + featBytes);                  // [N,256], becomes z in place
    float* deg   = (float*)(ws + 2 * featBytes);              // [N] deg -> dinv in place
    float* stats = (float*)(ws + 2 * featBytes + (size_t)N * sizeof(float));          // [512]
    float* bnp   = (float*)(ws + 2 * featBytes + (size_t)N * sizeof(float) + 2048);   // [512]

    // 1) init deg=1 (self loop), stats=0
    {
        int n = (N > 2 * COUT) ? N : 2 * COUT;
        k_init<<<(n + 255) / 256, 256, 0, stream>>>(deg, stats, N);
    }
    // 2) degree over edges
    k_degree<<<(E + 255) / 256, 256, 0, stream>>>(edst, deg, E);
    // 3) dinv = rsqrt(deg)
    k_dinv<<<(N + 255) / 256, 256, 0, stream>>>(deg, N);
    // 4) y = dinv * (x @ W^T) via f32 WMMA
    k_gemm_wmma<<<(N + 15) / 16, 512, 0, stream>>>(x, W, deg, y, N);
    // 5) agg = y (self-loop term)
    {
        long n4 = (long)N * COUT / 4;
        k_copy4<<<(unsigned)((n4 + 255) / 256), 256, 0, stream>>>((const float4*)y, (float4*)agg, n4);
    }
    // 6) scatter-add over edges
    {
        long tot = (long)E * 64;
        k_scatter<<<(unsigned)((tot + 255) / 256), 256, 0, stream>>>(esrc, edst, y, agg, E);
    }
    // 7) z = relu(dinv*agg + b) in place + BN stats
    k_finalize<<<(N + ROWS_PER_BLK - 1) / ROWS_PER_BLK, 256, 0, stream>>>(agg, deg, bias, stats, N);
    // 8) BN scale/shift
    k_bnparam<<<1, 256, 0, stream>>>(stats, gamma, beta, bnp, N);
    // 9) BN affine + maxpool pairs -> out [N,128]
    {
        long tot = (long)N * CHALF;
        k_out<<<(unsigned)((tot + 255) / 256), 256, 0, stream>>>(agg, bnp, out, N);
    }
}